// VoxelProjection_fish_78245714198736
// MI455X (gfx1250) — compile-verified
//
#include <hip/hip_runtime.h>

// VoxelProjection_fish: weighted gather from a fisheye feature map into a BEV
// grid.  Pure memory-bound kernel (232 MB of stores, L2-resident gathers).
// CDNA5 paths used: GLOBAL_LOAD_ASYNC_TO_LDS_B128 (+ s_wait_asynccnt) to stage
// per-block voxel metadata, global_prefetch_b8 for the random gathers, and a
// coalesced B128 store stream for the output.

namespace {
constexpr int kCIn   = 336;
constexpr int kZ     = 6;
constexpr int kBevH  = 240;
constexpr int kBevW  = 120;
constexpr int kHWBev = kBevH * kBevW;      // 28800 spatial cells per z-slab
constexpr int kL     = kZ * kHWBev;        // 172800 voxels
constexpr int kHF    = 160;
constexpr int kWF    = 256;
constexpr int kPix   = kHF * kWF;          // 40960 pixels per channel plane

constexpr int kVoxPerThread = 4;           // one float4 store per channel
constexpr int kCChunks      = 8;           // channel split for occupancy
constexpr int kCPerChunk    = kCIn / kCChunks;  // 42
constexpr int kThreads      = 256;         // 8 wave32s per workgroup
constexpr int kVoxPerBlock  = kThreads * kVoxPerThread;  // 1024
constexpr int kPrefetchDist = 6;           // channels ahead to prefetch
}  // namespace

__global__ __launch_bounds__(kThreads)
void voxproj_fish_kernel(const float* __restrict__ input,
                         const int*   __restrict__ uu,
                         const int*   __restrict__ vv,
                         const float* __restrict__ valid,
                         const float* __restrict__ density,
                         float*       __restrict__ out)
{
    // Block-local staging buffers for the metadata slice (4 x 4 KB = 16 KB).
    __shared__ __align__(16) int   s_u[kVoxPerBlock];
    __shared__ __align__(16) int   s_v[kVoxPerBlock];
    __shared__ __align__(16) float s_a[kVoxPerBlock];
    __shared__ __align__(16) float s_d[kVoxPerBlock];

    const int tib = threadIdx.x;
    const int t   = blockIdx.x * kThreads + tib;
    const int l0  = t * kVoxPerThread;
    if (l0 >= kL) return;

    // ---- Async-copy this lane's 16B chunk of each metadata array into LDS.
    // GVS addressing: mem = SADDR(64b) + VADDR(32b byte offset).  VDST holds
    // the per-lane LDS byte offset (low 32 bits of the generic pointer).
    {
        const unsigned goff  = (unsigned)l0 * 4u;  // byte offset into arrays
        const unsigned lds_u = (unsigned)(size_t)(&s_u[tib * kVoxPerThread]);
        const unsigned lds_v = (unsigned)(size_t)(&s_v[tib * kVoxPerThread]);
        const unsigned lds_a = (unsigned)(size_t)(&s_a[tib * kVoxPerThread]);
        const unsigned lds_d = (unsigned)(size_t)(&s_d[tib * kVoxPerThread]);
        asm volatile(
            "global_load_async_to_lds_b128 %0, %4, %5\n\t"
            "global_load_async_to_lds_b128 %1, %4, %6\n\t"
            "global_load_async_to_lds_b128 %2, %4, %7\n\t"
            "global_load_async_to_lds_b128 %3, %4, %8\n\t"
            "s_wait_asynccnt 0x0"
            :
            : "v"(lds_u), "v"(lds_v), "v"(lds_a), "v"(lds_d),
              "v"(goff),
              "s"(uu), "s"(vv), "s"(valid), "s"(density)
            : "memory");
    }

    // Each lane consumes exactly the chunk it copied -> per-wave
    // s_wait_asynccnt above is sufficient, no workgroup barrier needed.
    const int g  = tib * kVoxPerThread;
    const int i0 = s_v[g + 0] * kWF + s_u[g + 0];
    const int i1 = s_v[g + 1] * kWF + s_u[g + 1];
    const int i2 = s_v[g + 2] * kWF + s_u[g + 2];
    const int i3 = s_v[g + 3] * kWF + s_u[g + 3];
    const float w0 = s_a[g + 0] * s_d[g + 0];
    const float w1 = s_a[g + 1] * s_d[g + 1];
    const float w2 = s_a[g + 2] * s_d[g + 2];
    const float w3 = s_a[g + 3] * s_d[g + 3];

    const int c0 = blockIdx.y * kCPerChunk;

    // Aligned groups of 4 voxels never cross a z-slab (28800 % 4 == 0).
    const int z = l0 / kHWBev;
    const int s = l0 - z * kHWBev;

    const float* __restrict__ src = input + (size_t)c0 * kPix;
    float* __restrict__ dst =
        out + (size_t)(z * kCIn + c0) * kHWBev + s;

#pragma unroll 6
    for (int c = 0; c < kCPerChunk; ++c) {
        const float* __restrict__ p = src + (size_t)c * kPix;

        // Pull the upcoming channel's gather lines toward the WGP
        // (lowers to global_prefetch_b8 on gfx1250).
        if (c + kPrefetchDist < kCPerChunk) {
            __builtin_prefetch(p + (size_t)kPrefetchDist * kPix + i0, 0, 0);
        }

        float4 r;
        r.x = p[i0] * w0;
        r.y = p[i1] * w1;
        r.z = p[i2] * w2;
        r.w = p[i3] * w3;
        // Lane-contiguous along s -> fully coalesced 128B store transactions.
        *reinterpret_cast<float4*>(dst + (size_t)c * kHWBev) = r;
    }
}

extern "C" void kernel_launch(void* const* d_in, const int* in_sizes, int n_in,
                              void* d_out, int out_size, void* d_ws, size_t ws_size,
                              hipStream_t stream) {
    (void)in_sizes; (void)n_in; (void)out_size; (void)d_ws; (void)ws_size;

    const float* input   = (const float*)d_in[0];
    const int*   uu      = (const int*)d_in[1];
    const int*   vv      = (const int*)d_in[2];
    const float* valid   = (const float*)d_in[3];
    const float* density = (const float*)d_in[4];
    float*       out     = (float*)d_out;

    const int vox_groups = kL / kVoxPerThread;                 // 43200
    dim3 grid((vox_groups + kThreads - 1) / kThreads,          // 169
              kCChunks);                                       // 8
    dim3 block(kThreads, 1, 1);

    voxproj_fish_kernel<<<grid, block, 0, stream>>>(input, uu, vv, valid,
                                                    density, out);
}